// TwoBranchDGridConv_14242111554212
// MI455X (gfx1250) — compile-verified
//
#include <hip/hip_runtime.h>

// Problem constants (from reference): B=512, CIN=512, COUT=512, K=3, H=W=5
#define NBATCH 512
#define CIN    512
#define COUT   512
#define KS     3
#define KK     9            // K*K
#define HW     25           // H*W
#define CK     (CIN * KK)   // 4608  (per-branch K extent)
#define KT     64           // K-chunk staged in LDS per iteration
#define NB     2            // batches per workgroup
#define NCOL   (NB * 32)    // 64 N columns (25 valid + 7 pad per batch)
#define NTILES (NCOL / 16)  // 4 N-tiles
#define BSTRIDE 72          // LDS row stride for B chunk (b64-aligned, bank-staggered)
#define NTHREADS 512        // 16 wave32
#define NCHUNK_PER_BRANCH (CK / KT)          // 72
#define NCHUNK (2 * NCHUNK_PER_BRANCH)       // 144

typedef float v2f __attribute__((ext_vector_type(2)));
typedef float v8f __attribute__((ext_vector_type(8)));

__launch_bounds__(NTHREADS)
__global__ void dgrid_conv_wmma(const float* __restrict__ x,
                                const float* __restrict__ atten,
                                const float* __restrict__ w1,
                                const float* __restrict__ w2,
                                float* __restrict__ out)
{
    // LDS (~124 KB): leans on CDNA5's 320 KB-per-WGP LDS; 2 WGs still fit a WGP.
    __shared__ __align__(16) float xs[NB * CIN * HW];    // 102400 B
    __shared__ float ats[NB * KK * HW];                  //   1800 B
    __shared__ __align__(16) float Bs[NCOL * BSTRIDE];   //  18432 B
    __shared__ int zmap[KK * HW];                        //    900 B (zero-pad gather, -1 = OOB)
    __shared__ int cmap[KK * HW];                        //    900 B (circular gather)

    const int tid   = threadIdx.x;
    const int wave  = tid >> 5;       // 0..15
    const int lane  = tid & 31;
    const int lhalf = lane >> 4;      // 0/1
    const int l16   = lane & 15;
    const int kb    = lhalf * 2;      // K offset of this lane's fragment pair (0 or 2)

    const int bbase = blockIdx.x * NB;   // two batches per workgroup

    // ---- stage x[bbase..bbase+1] into LDS with float4 loads ----
    {
        const float4* xg  = (const float4*)(x + (size_t)bbase * CIN * HW);
        float4*       xs4 = (float4*)xs;
        for (int i = tid; i < (NB * CIN * HW) / 4; i += NTHREADS) xs4[i] = xg[i];
    }
    // ---- stage atten ----
    {
        const float* ag = atten + (size_t)bbase * KK * HW;
        for (int i = tid; i < NB * KK * HW; i += NTHREADS) ats[i] = ag[i];
    }
    // ---- gather maps: (k,l) -> pixel index in 5x5 image ----
    if (tid < KK * HW) {
        int k  = tid / HW, l = tid % HW;
        int ki = k / KS,  kj = k % KS;
        int r  = l / 5 + ki - 1;
        int c  = l % 5 + kj - 1;
        zmap[tid] = (r >= 0 && r < 5 && c >= 0 && c < 5) ? (r * 5 + c) : -1;
        int rw = (r + 5) % 5, cw = (c + 5) % 5;
        cmap[tid] = rw * 5 + cw;
    }
    __syncthreads();

    // Accumulators: wave covers M rows [wave*32, wave*32+31] x all 64 N cols.
    v8f acc[2][NTILES];
    const v8f vzero = {0.f, 0.f, 0.f, 0.f, 0.f, 0.f, 0.f, 0.f};
#pragma unroll
    for (int mt = 0; mt < 2; ++mt)
#pragma unroll
        for (int nt = 0; nt < NTILES; ++nt) acc[mt][nt] = vzero;

    // Build-phase work split: each thread fills 8 K-rows of one of 64 columns.
    const int bn     = tid >> 3;        // column 0..63
    const int bl     = bn >> 5;         // local batch 0/1
    const int bcol   = bn & 31;         // l within batch (0..31)
    const int brbase = (tid & 7) * 8;   // K-row base within chunk

    for (int chunk = 0; chunk < NCHUNK; ++chunk) {
        const int    br    = (chunk >= NCHUNK_PER_BRANCH);
        const int    kbase = (br ? chunk - NCHUNK_PER_BRANCH : chunk) * KT;
        const float* wsel  = br ? w2 : w1;

        // ---- cooperative build of B chunk: Bs[col][k] = atten * gathered patch ----
        __syncthreads();   // previous chunk's readers done
        if (bcol < HW) {
#pragma unroll
            for (int rr = 0; rr < 8; ++rr) {
                int      r    = brbase + rr;
                unsigned kloc = (unsigned)(kbase + r);
                unsigned c    = kloc / 9u;             // channel
                int      k    = (int)(kloc - c * 9u);  // kernel tap
                int      midx = k * HW + bcol;
                int      idx  = br ? cmap[midx] : zmap[midx];
                float    xv   = (idx >= 0) ? xs[bl * (CIN * HW) + c * HW + (unsigned)idx] : 0.0f;
                Bs[bn * BSTRIDE + r] = ats[bl * (KK * HW) + midx] * xv;
            }
        } else if (chunk == 0) {       // pad columns stay zero forever
#pragma unroll
            for (int rr = 0; rr < 8; ++rr) Bs[bn * BSTRIDE + brbase + rr] = 0.0f;
        }
        __syncthreads();

        // Prefetch next chunk's weight rows (global_prefetch_b8 path).
        if (chunk + 1 < NCHUNK) {
            const int    br2    = (chunk + 1 >= NCHUNK_PER_BRANCH);
            const int    kbase2 = (br2 ? chunk + 1 - NCHUNK_PER_BRANCH : chunk + 1) * KT;
            const float* wsel2  = br2 ? w2 : w1;
#pragma unroll
            for (int mt = 0; mt < 2; ++mt) {
                int m = wave * 32 + mt * 16 + l16;
                __builtin_prefetch(wsel2 + (size_t)m * CK + kbase2, 0, 0);
            }
        }

        // ---- compute: 16 K-steps of V_WMMA_F32_16X16X4_F32 ----
#pragma unroll
        for (int s = 0; s < 16; ++s) {
            const int kk = s * 4;
            v2f afr[2], bfr[NTILES];
#pragma unroll
            for (int mt = 0; mt < 2; ++mt) {
                int m = wave * 32 + mt * 16 + l16;   // A: row = lane%16, K pair at kb
                afr[mt] = *(const v2f*)(wsel + (size_t)m * CK + (kbase + kk + kb));
            }
#pragma unroll
            for (int nt = 0; nt < NTILES; ++nt) {
                int col = nt * 16 + l16;             // B: col = lane%16, K pair at kb
                bfr[nt] = *(const v2f*)(&Bs[col * BSTRIDE + kk + kb]);
            }
#pragma unroll
            for (int mt = 0; mt < 2; ++mt)
#pragma unroll
                for (int nt = 0; nt < NTILES; ++nt)
                    acc[mt][nt] = __builtin_amdgcn_wmma_f32_16x16x4_f32(
                        false, afr[mt], false, bfr[nt],
                        (short)0, acc[mt][nt], false, false);
        }
    }

    // ---- store D tiles: VGPR v holds rows v (lanes 0-15) / v+8 (lanes 16-31) ----
#pragma unroll
    for (int nt = 0; nt < NTILES; ++nt) {
        const int col = nt * 16 + l16;
        const int blc = col >> 5;        // local batch
        const int l   = col & 31;
        if (l < HW) {
#pragma unroll
            for (int mt = 0; mt < 2; ++mt) {
                const int obase = wave * 32 + mt * 16 + lhalf * 8;
                float* op = out + ((size_t)(bbase + blc) * COUT + obase) * HW + l;
#pragma unroll
                for (int v = 0; v < 8; ++v)
                    op[(size_t)v * HW] = acc[mt][nt][v];
            }
        }
    }
}

extern "C" void kernel_launch(void* const* d_in, const int* in_sizes, int n_in,
                              void* d_out, int out_size, void* d_ws, size_t ws_size,
                              hipStream_t stream) {
    const float* x     = (const float*)d_in[0];
    const float* atten = (const float*)d_in[1];
    const float* w1    = (const float*)d_in[2];
    const float* w2    = (const float*)d_in[3];
    float*       out   = (float*)d_out;

    dim3 grid(NBATCH / NB);   // 256 workgroups, two batches each
    dim3 block(NTHREADS);     // 16 wave32
    hipLaunchKernelGGL(dgrid_conv_wmma, grid, block, 0, stream, x, atten, w1, w2, out);
}